// ClusterMemory_85126251807521
// MI455X (gfx1250) — compile-verified
//
#include <hip/hip_runtime.h>
#include <hip/hip_bf16.h>

typedef __attribute__((ext_vector_type(2))) float v2f;
typedef __attribute__((ext_vector_type(8))) float v8f;
typedef __attribute__((ext_vector_type(4))) int   v4i;

#define Bsz 128
#define Dd  768
#define Nn  65536
#define KC  32      // K chunk staged in LDS
#define LDB 44      // padded LDS row stride (dwords): 176B rows -> 16B-aligned, conflict-free
#define NT  4       // 16-wide N tiles per workgroup (64 columns)
#define NCHUNK (Dd / KC)

#define GLOBAL_AS __attribute__((address_space(1)))
#define LDS_AS    __attribute__((address_space(3)))

#if defined(__AMDGCN__) && __has_builtin(__builtin_amdgcn_global_load_async_to_lds_b128)
#define USE_ASYNC_LDS 1
#else
#define USE_ASYNC_LDS 0
#endif

__device__ __forceinline__ void async_copy_wait() {
#if USE_ASYNC_LDS
#if __has_builtin(__builtin_amdgcn_s_wait_asynccnt)
    __builtin_amdgcn_s_wait_asynccnt(0);
#else
    asm volatile("s_wait_asynccnt 0x0" ::: "memory");
#endif
#endif
}

__device__ __forceinline__ void stage_b128(const float* src, float* dst) {
#if USE_ASYNC_LDS
    // cpol = 1 -> TH=NT: features are a read-once 201MB stream; keep L2 for A
    __builtin_amdgcn_global_load_async_to_lds_b128(
        (GLOBAL_AS v4i*)src, (LDS_AS v4i*)dst, 0, 1);
#else
    const float4 d = *reinterpret_cast<const float4*>(src);
    dst[0] = d.x; dst[1] = d.y; dst[2] = d.z; dst[3] = d.w;
#endif
}

// ---------------------------------------------------------------------------
// 1) Row-normalize i_feats -> A  (A[b][d] = i_feats[b][d] / max(||row||, eps))
// ---------------------------------------------------------------------------
__global__ __launch_bounds__(256) void normalize_kernel(const float* __restrict__ in,
                                                        float* __restrict__ outp) {
    __shared__ float red[256];
    const int b = blockIdx.x, t = threadIdx.x;
    const float* row = in + (size_t)b * Dd;
    float s = 0.0f;
    for (int j = t; j < Dd; j += 256) { float v = row[j]; s += v * v; }
    red[t] = s; __syncthreads();
    for (int st = 128; st > 0; st >>= 1) {
        if (t < st) red[t] += red[t + st];
        __syncthreads();
    }
    const float inv = 1.0f / fmaxf(sqrtf(red[0]), 1e-12f);
    for (int j = t; j < Dd; j += 256) outp[(size_t)b * Dd + j] = row[j] * inv;
}

// ---------------------------------------------------------------------------
// 2) WMMA GEMM:  out[b][n] = sum_k A[b][k] * feat[rowidx(n)][k]
//    A is WMMA-A (16x4 f32), feature tile is WMMA-B (4x16 f32), staged via a
//    double-buffered LDS tile filled with GLOBAL_LOAD_ASYNC_TO_LDS_B128 (NT).
//    gather == nullptr -> rowidx(n) = n (memory-bank GEMM, ncols = 65536)
//    gather != nullptr -> rowidx(n) = gather[n] (outputs1, ncols = 128)
// ---------------------------------------------------------------------------
__global__ __launch_bounds__(256) void gemm_wmma_kernel(const float* __restrict__ A,
                                                        const float* __restrict__ feat,
                                                        const int* __restrict__ gather,
                                                        float* __restrict__ out,
                                                        int ncols) {
    __shared__ float lb[2][64 * LDB];
    __shared__ int   ridx[64];

    const int tid  = threadIdx.x;
    const int wave = tid >> 5;
    const int lane = tid & 31;
    const int lrow = lane & 15;            // row-within-tile for A, col for B/D
    const int lkhi = (lane >> 4) << 1;     // K sub-offset: 0 (lanes 0-15) or 2 (16-31)
    const int n_base = blockIdx.x * (NT * 16);
    const int m_base = wave * 16;          // 8 waves cover all 128 batch rows

    if (tid < 64) {
        const int n = n_base + tid;
        ridx[tid] = gather ? gather[n] : n;
    }
    __syncthreads();

    // Each thread owns two float4 slots of the 64x32 tile; hoist the global
    // source pointers (advance by KC floats/chunk) and the 4 LDS dst pointers.
    const int r0 = tid >> 3,         c40 = (tid & 7) * 4;
    const int r1 = (256 + tid) >> 3, c41 = c40;          // (256+tid)&7 == tid&7
    const float* s0 = feat + (size_t)ridx[r0] * Dd + c40;
    const float* s1 = feat + (size_t)ridx[r1] * Dd + c41;
    float* const d00 = &lb[0][r0 * LDB + c40];
    float* const d01 = &lb[0][r1 * LDB + c41];
    float* const d10 = &lb[1][r0 * LDB + c40];
    float* const d11 = &lb[1][r1 * LDB + c41];

    v8f acc[NT] = {};
    const float* arow = A + (size_t)(m_base + lrow) * Dd;

    // prime buffer 0 with chunk 0
    stage_b128(s0, d00);
    stage_b128(s1, d01);
    s0 += KC; s1 += KC;

    for (int c = 0; c < NCHUNK; ++c) {
        async_copy_wait();      // our async writes for buf[c&1] have landed
        __syncthreads();        // everyone's writes landed; prev buffer fully read

        if (c + 1 < NCHUNK) {
            if ((c + 1) & 1) { stage_b128(s0, d10); stage_b128(s1, d11); }
            else             { stage_b128(s0, d00); stage_b128(s1, d01); }
            s0 += KC; s1 += KC;
        }

        const int    k0  = c * KC;
        const float* lbc = lb[c & 1];
        #pragma unroll
        for (int kk = 0; kk < KC / 4; ++kk) {
            // A fragment: lane l -> A[m_base + l%16][k0 + 4*kk + lkhi + {0,1}]
            const v2f a = *reinterpret_cast<const v2f*>(arow + k0 + kk * 4 + lkhi);
            #pragma unroll
            for (int t = 0; t < NT; ++t) {
                // B fragment: lane l, vgpr r -> B[k = r + lkhi][n = l%16]
                const v2f b = *reinterpret_cast<const v2f*>(
                    &lbc[(t * 16 + lrow) * LDB + kk * 4 + lkhi]);
                acc[t] = __builtin_amdgcn_wmma_f32_16x16x4_f32(
                    false, a, false, b, (short)0, acc[t], false, false);
            }
        }
        // no trailing barrier: next iteration's barrier is the WAR guard
    }

    // D layout: lane l, vgpr r -> D[m = r + 8*(l/16)][n = l%16]
    const int rowb = m_base + ((lane >> 4) << 3);
    #pragma unroll
    for (int t = 0; t < NT; ++t) {
        const int col = n_base + t * 16 + lrow;
        #pragma unroll
        for (int r = 0; r < 8; ++r) {
            __builtin_nontemporal_store(acc[t][r],
                                        &out[(size_t)(rowb + r) * ncols + col]);
        }
    }
}

// ---------------------------------------------------------------------------
// 3) Loss: soft-triplet on outputs1 [128x128] + 0.08 * mean(1 - outputs1)
// ---------------------------------------------------------------------------
__global__ __launch_bounds__(128) void loss_kernel(const float* __restrict__ s,
                                                   const int* __restrict__ tgt,
                                                   float* __restrict__ out) {
    __shared__ float red[128];
    __shared__ int   tsh[128];
    __shared__ float tri_s;
    const int i = threadIdx.x;
    tsh[i] = tgt[i];
    __syncthreads();
    const int ti = tsh[i];
    const float* row = s + (size_t)i * Bsz;
    const float tau_inv = 50.0f;  // 1/0.02

    float pmax = -1e30f, nmax = -1e30f, csum = 0.0f;
    bool  has_neg = false;
    for (int j = 0; j < Bsz; ++j) {
        const float v = row[j];
        csum += 1.0f - v;
        if (tsh[j] == ti) {
            pmax = fmaxf(pmax, -v * tau_inv);
        } else {
            has_neg = true;
            nmax = fmaxf(nmax, v * tau_inv);
        }
    }
    float pe = 0.0f, pw = 0.0f, ne = 0.0f, nw = 0.0f;
    for (int j = 0; j < Bsz; ++j) {
        const float v = row[j];
        if (tsh[j] == ti) {
            const float e = __expf(-v * tau_inv - pmax);
            pe += e; pw += e * v;
        } else {
            const float e = __expf(v * tau_inv - nmax);
            ne += e; nw += e * v;
        }
    }
    const float pos_v = pw / fmaxf(pe, 1e-12f);
    const float neg_v = nw / fmaxf(ne, 1e-12f);
    float li = fmaxf(neg_v - pos_v + 0.1f, 0.0f);
    if (!has_neg) li = 0.0f;

    red[i] = li; __syncthreads();
    for (int st = 64; st > 0; st >>= 1) {
        if (i < st) red[i] += red[i + st];
        __syncthreads();
    }
    if (i == 0) tri_s = red[0] * (1.0f / 128.0f);
    __syncthreads();

    red[i] = csum; __syncthreads();
    for (int st = 64; st > 0; st >>= 1) {
        if (i < st) red[i] += red[i + st];
        __syncthreads();
    }
    if (i == 0) {
        const float center = red[0] * (1.0f / (128.0f * 128.0f));
        out[0] = tri_s + 0.08f * center;
    }
}

// ---------------------------------------------------------------------------
extern "C" void kernel_launch(void* const* d_in, const int* in_sizes, int n_in,
                              void* d_out, int out_size, void* d_ws, size_t ws_size,
                              hipStream_t stream) {
    const float* i_feats  = (const float*)d_in[0];   // [128, 768]
    const int*   targets  = (const int*)d_in[1];     // [128]
    const float* features = (const float*)d_in[2];   // [65536, 768]
    float* out = (float*)d_out;                      // [1 + 128*65536]

    float* ws         = (float*)d_ws;
    float* inputs_nrm = ws;                  // 128*768 floats
    float* outputs1   = ws + Bsz * Dd;       // 128*128 floats

    normalize_kernel<<<Bsz, 256, 0, stream>>>(i_feats, inputs_nrm);

    // big memory-bank GEMM -> d_out + 1  (streams 201 MB of features once)
    gemm_wmma_kernel<<<Nn / (NT * 16), 256, 0, stream>>>(
        inputs_nrm, features, nullptr, out + 1, Nn);

    // gathered 128x128 GEMM -> outputs1
    gemm_wmma_kernel<<<Bsz / (NT * 16), 256, 0, stream>>>(
        inputs_nrm, features, targets, outputs1, Bsz);

    loss_kernel<<<1, Bsz, 0, stream>>>(outputs1, targets, out);
}